// SelfAttentionBlock_5944234738243
// MI455X (gfx1250) — compile-verified
//
#include <hip/hip_runtime.h>
#include <hip/hip_bf16.h>

#define B_DIM   2
#define C_DIM   128
#define S_DIM   1024
#define GROUPS  32
#define GSIZE   4          // channels per group (C/GROUPS)
#define EPS_GN  1e-5f

typedef __attribute__((ext_vector_type(2))) float v2f;
typedef __attribute__((ext_vector_type(8))) float v8f;

// ---------------------------------------------------------------------------
// Kernel 1: GroupNorm over (4 channels x 1024 spatial) per group, then write
// the normalized activations transposed to hbuf[B*S, C] (row-major GEMM input).
// One 256-thread block per (b, g): 64 blocks.
// ---------------------------------------------------------------------------
__global__ void gn_transpose_kernel(const float* __restrict__ x,   // [B, C, S]
                                    const float* __restrict__ gw,  // [C]
                                    const float* __restrict__ gb,  // [C]
                                    float* __restrict__ hbuf) {    // [B*S, C]
    __shared__ float r_sum[256];
    __shared__ float r_sq[256];
    __shared__ float s_mean, s_rstd;

    const int b   = blockIdx.x / GROUPS;
    const int g   = blockIdx.x % GROUPS;
    const int tid = threadIdx.x;
    const float* xg = x + (b * C_DIM + g * GSIZE) * S_DIM;  // 4 x 1024 block

    float sum = 0.f, sq = 0.f;
    for (int e = tid; e < GSIZE * S_DIM; e += 256) {
        float v = xg[e];
        sum += v;
        sq  += v * v;
    }
    r_sum[tid] = sum;
    r_sq[tid]  = sq;
    __syncthreads();
    for (int off = 128; off > 0; off >>= 1) {
        if (tid < off) {
            r_sum[tid] += r_sum[tid + off];
            r_sq[tid]  += r_sq[tid + off];
        }
        __syncthreads();
    }
    if (tid == 0) {
        const float inv_n = 1.0f / (GSIZE * S_DIM);
        float m   = r_sum[0] * inv_n;
        float var = r_sq[0] * inv_n - m * m;      // biased var (matches jnp.var)
        s_mean = m;
        s_rstd = rsqrtf(var + EPS_GN);
    }
    __syncthreads();
    const float m = s_mean, r = s_rstd;
    for (int e = tid; e < GSIZE * S_DIM; e += 256) {
        int cc = e >> 10;           // e / 1024
        int s  = e & 1023;          // e % 1024
        int c  = g * GSIZE + cc;
        float v = (xg[e] - m) * r * gw[c] + gb[c];
        hbuf[(b * S_DIM + s) * C_DIM + c] = v;
    }
}

// ---------------------------------------------------------------------------
// Kernel 2: fp32 GEMM via V_WMMA_F32_16X16X4_F32.
//   out[m][n] = sum_k A[m][k] * W[n][k] + bias[n]   (m = b*S+s)
// written transposed into [B, C, S] layout, with optional residual add.
// Grid: (B*S)/16 blocks of 256 threads; wave w handles column tile w (8x16=128=C).
//
// Fragment layout (ISA 7.12.2, 32-bit, wave32):
//   A 16x4 : lanes 0-15 row M=lane, VGPR{0,1}=K{0,1}; lanes 16-31: K{2,3}
//   B 4x16 : lanes 0-15 col N=lane, VGPR{0,1}=K{0,1}; lanes 16-31: K{2,3}
//   -> each lane loads one float2 (consecutive K) from one row of A / W.
//   D 16x16: VGPR r -> M = r (lanes 0-15) / r+8 (lanes 16-31), N = lane%16.
// ---------------------------------------------------------------------------
__global__ void gemm_wmma_kernel(const float* __restrict__ A,     // [B*S, C]
                                 const float* __restrict__ W,     // [C, C]
                                 const float* __restrict__ bias,  // [C]
                                 const float* __restrict__ resid, // [B, C, S] or nullptr
                                 float* __restrict__ out) {       // [B, C, S]
    const int lane = threadIdx.x & 31;
    const int wave = threadIdx.x >> 5;
    const int row0 = blockIdx.x << 4;     // M tile base (b*S+s)
    const int col0 = wave << 4;           // N tile base (output channel)
    const int half = lane >> 4;           // 0: K 0,1   1: K 2,3
    const int lr   = lane & 15;

    const float* arow = A + (size_t)(row0 + lr) * C_DIM + 2 * half;
    const float* brow = W + (size_t)(col0 + lr) * C_DIM + 2 * half;

    v8f acc = {};
#pragma unroll
    for (int k0 = 0; k0 < C_DIM; k0 += 4) {
        v2f afrag = *(const v2f*)(arow + k0);
        v2f bfrag = *(const v2f*)(brow + k0);
        acc = __builtin_amdgcn_wmma_f32_16x16x4_f32(
            /*neg_a=*/false, afrag, /*neg_b=*/false, bfrag,
            /*c_mod=*/(short)0, acc, /*reuse_a=*/false, /*reuse_b=*/false);
    }

    const int   n  = col0 + lr;
    const float bb = bias[n];
#pragma unroll
    for (int r = 0; r < 8; ++r) {
        int m = row0 + r + half * 8;      // global row index (b*S+s)
        int bi = m >> 10;                 // m / 1024
        int s  = m & 1023;                // m % 1024
        int idx = (bi * C_DIM + n) * S_DIM + s;
        float val = acc[r] + bb;
        if (resid) val += resid[idx];
        out[idx] = val;
    }
}

// ---------------------------------------------------------------------------
// Kernel 3: HEAD_DIM=1 attention. Each (b, head c) is an independent problem:
//   out_i = sum_j exp(q_i*k_j - m_i) * v_j / sum_j exp(q_i*k_j - m_i)
// with exact m_i = q_i>=0 ? q_i*max(k) : q_i*min(k).
// One 256-thread block per (b,c) = 256 blocks.
//
// k and v rows (4 KB each) are staged into LDS with the CDNA5 async copy
// engine: one GLOBAL_LOAD_ASYNC_TO_LDS_B128 per lane per buffer
// (256 lanes x 16 B = 4 KB), tracked by ASYNCcnt. Each wave drains its own
// ASYNCcnt, then the workgroup barrier makes all LDS writes visible.
// ---------------------------------------------------------------------------
__global__ void attn_hd1_kernel(const float* __restrict__ q,   // [B, C, S]
                                const float* __restrict__ k,   // [B, C, S]
                                const float* __restrict__ v,   // [B, C, S]
                                float* __restrict__ ao) {      // [B*S, C]
    __shared__ float ks[S_DIM];
    __shared__ float vs[S_DIM];
    __shared__ float rmx[256];
    __shared__ float rmn[256];

    const int b   = blockIdx.x >> 7;           // / 128
    const int c   = blockIdx.x & (C_DIM - 1);
    const int tid = threadIdx.x;

    const float* kp = k + (size_t)(b * C_DIM + c) * S_DIM;
    const float* vp = v + (size_t)(b * C_DIM + c) * S_DIM;

    // --- async global -> LDS staging (16 bytes per lane per buffer) ---
    {
        // Low 32 bits of the generic address of a __shared__ object are the
        // workgroup-relative LDS byte offset (aperture mapping, ISA 10.2).
        unsigned lk = (unsigned)(uintptr_t)(&ks[0]) + (unsigned)tid * 16u;
        unsigned lv = (unsigned)(uintptr_t)(&vs[0]) + (unsigned)tid * 16u;
        const float* gk = kp + tid * 4;
        const float* gv = vp + tid * 4;
        asm volatile("global_load_async_to_lds_b128 %0, %1, off"
                     :: "v"(lk), "v"(gk) : "memory");
        asm volatile("global_load_async_to_lds_b128 %0, %1, off"
                     :: "v"(lv), "v"(gv) : "memory");
        asm volatile("s_wait_asynccnt 0x0" ::: "memory");
    }
    __syncthreads();

    // --- exact row max/min of k for a safe softmax shift ---
    float kmx = -3.402823466e38f, kmn = 3.402823466e38f;
    for (int j = tid; j < S_DIM; j += 256) {
        float kv = ks[j];
        kmx = fmaxf(kmx, kv);
        kmn = fminf(kmn, kv);
    }
    rmx[tid] = kmx;
    rmn[tid] = kmn;
    __syncthreads();
    for (int off = 128; off > 0; off >>= 1) {
        if (tid < off) {
            rmx[tid] = fmaxf(rmx[tid], rmx[tid + off]);
            rmn[tid] = fminf(rmn[tid], rmn[tid + off]);
        }
        __syncthreads();
    }
    kmx = rmx[0];
    kmn = rmn[0];

    const float* qp = q + (size_t)(b * C_DIM + c) * S_DIM;
    for (int i = tid; i < S_DIM; i += 256) {
        const float qi = qp[i];
        const float mx = (qi >= 0.f) ? qi * kmx : qi * kmn;
        float se = 0.f, sv = 0.f;
        for (int j = 0; j < S_DIM; ++j) {
            float e = __expf(qi * ks[j] - mx);
            se += e;
            sv += e * vs[j];
        }
        ao[(size_t)(b * S_DIM + i) * C_DIM + c] = sv / se;
    }
}

// ---------------------------------------------------------------------------
// Host-side launcher. Inputs (setup_inputs order):
//  0 hidden_states [2,128,32,32]  1 gn_weight [128]  2 gn_bias [128]
//  3 Wq [128,128]  4 bq  5 Wk  6 bk  7 Wv  8 bv  9 Wo  10 bo
// Workspace: hbuf | q | k | v | ao, each B*S*C = 262144 floats (5 MB total).
// ---------------------------------------------------------------------------
extern "C" void kernel_launch(void* const* d_in, const int* in_sizes, int n_in,
                              void* d_out, int out_size, void* d_ws, size_t ws_size,
                              hipStream_t stream) {
    const float* x  = (const float*)d_in[0];
    const float* gw = (const float*)d_in[1];
    const float* gb = (const float*)d_in[2];
    const float* Wq = (const float*)d_in[3];
    const float* bq = (const float*)d_in[4];
    const float* Wk = (const float*)d_in[5];
    const float* bk = (const float*)d_in[6];
    const float* Wv = (const float*)d_in[7];
    const float* bv = (const float*)d_in[8];
    const float* Wo = (const float*)d_in[9];
    const float* bo = (const float*)d_in[10];
    float* out = (float*)d_out;

    float* ws = (float*)d_ws;
    const size_t NBS = (size_t)B_DIM * S_DIM * C_DIM;  // 262144
    float* hbuf = ws;
    float* qb   = ws + 1 * NBS;
    float* kb   = ws + 2 * NBS;
    float* vb   = ws + 3 * NBS;
    float* aob  = ws + 4 * NBS;

    const int gemm_blocks = (B_DIM * S_DIM) / 16;   // 128

    gn_transpose_kernel<<<B_DIM * GROUPS, 256, 0, stream>>>(x, gw, gb, hbuf);

    gemm_wmma_kernel<<<gemm_blocks, 256, 0, stream>>>(hbuf, Wq, bq, nullptr, qb);
    gemm_wmma_kernel<<<gemm_blocks, 256, 0, stream>>>(hbuf, Wk, bk, nullptr, kb);
    gemm_wmma_kernel<<<gemm_blocks, 256, 0, stream>>>(hbuf, Wv, bv, nullptr, vb);

    attn_hd1_kernel<<<B_DIM * C_DIM, 256, 0, stream>>>(qb, kb, vb, aob);

    // Output projection + residual, written back to [B, C, S] (== [B,C,H,W]).
    gemm_wmma_kernel<<<gemm_blocks, 256, 0, stream>>>(aob, Wo, bo, x, out);
}